// HybridOHEMBCELoss_19146964206144
// MI455X (gfx1250) — compile-verified
//
#include <hip/hip_runtime.h>
#include <stdint.h>

// ---------------- problem geometry ----------------
#define N_CH0        (8 * 512 * 512)   // elements in the [:,0] slice = 2,097,152
#define BATCH_STRIDE (8 * 512 * 512)   // C*H*W (floats) between batches
#define HW           (512 * 512)       // contiguous channel-0 run per batch
#define CHUNK        4096              // floats per TDM tile (16 KB)
#define CHUNKS_PER_BATCH (HW / CHUNK)  // 64
#define NCHUNK       (N_CH0 / CHUNK)   // 512
#define K1_BLOCKS    256
#define CHUNKS_PER_BLOCK (NCHUNK / K1_BLOCKS) // 2
#define NBINS        1024
#define BIN_SCALE    16384.0f          // |x| histogram resolution for top-k fallback
#define LOGIT_07     0.847297860387204f // logit(0.7); sigmoid(x)<=0.7 <=> x<=LOGIT_07

struct WS {                     // workspace layout (zeroed each call)
  float    n_kept;              // 0
  unsigned flag;                // 4   (1 => fallback/top-k path active)
  double   sum_elem;            // 8
  double   sum_elem_u;          // 16
  float    n_union;             // 24
  float    thresh;              // 28
  unsigned hist[NBINS];         // 32..4128
};

typedef unsigned int u32x4 __attribute__((ext_vector_type(4)));
typedef int          i32x8 __attribute__((ext_vector_type(8)));
typedef int          i32x4 __attribute__((ext_vector_type(4)));

// ---- CDNA5 Tensor Data Mover: 1-D tile of `nelem` f32 global -> LDS ----
// D# per ISA cdna5 §8.3/8.4: group0 {count, lds_addr, global_addr, type=2},
// group1 {data_size=4B, tensor_dim0=nelem, tensor_dim1=1, tile_dim0=nelem,
//         tile_dim1=0 (1-D tile), dim0_stride=nelem}. Groups 2/3 unused (<=2D).
__device__ __forceinline__ void tdm_load_f32(const float* gsrc, void* lds, unsigned nelem) {
  unsigned long long ga = (unsigned long long)(uintptr_t)gsrc;
  unsigned lds_off = (unsigned)(uintptr_t)lds;   // low 32b of generic ptr == LDS byte addr
  u32x4 g0;
  g0[0] = 1u;                                            // count=1, user descriptor
  g0[1] = lds_off;                                       // lds_addr [63:32]
  g0[2] = (unsigned)ga;                                  // global_addr [95:64]
  g0[3] = (unsigned)((ga >> 32) & 0x1FFFFFFull)          // global_addr [120:96]
        | (2u << 30);                                    // type=2 ("image")
  i32x8 g1;
  g1[0] = (int)(2u << 16);                               // data_size=2 -> 4 bytes
  g1[1] = (int)((nelem & 0xFFFFu) << 16);                // tensor_dim0[15:0]
  g1[2] = (int)((nelem >> 16) | (1u << 16));             // tensor_dim0[31:16] | tensor_dim1=1
  g1[3] = (int)((nelem & 0xFFFFu) << 16);                // tile_dim0 = nelem
  g1[4] = 0;                                             // tile_dim1=0 (1-D), tile_dim2=0
  g1[5] = (int)nelem;                                    // tensor_dim0_stride[31:0]
  g1[6] = (int)((nelem & 0xFFFFu) << 16);                // dim1_stride[15:0] (unused)
  g1[7] = 0;
  i32x4 z4 = {0, 0, 0, 0};
  i32x8 z8 = {0, 0, 0, 0, 0, 0, 0, 0};
  // clang-23 / therock 6-arg form: (g0, g1, g2, g3, extra, cpol)
  __builtin_amdgcn_tensor_load_to_lds(g0, g1, z4, z4, z8, 0);
}

__device__ __forceinline__ float wave_red(float v) {
  #pragma unroll
  for (int o = 16; o > 0; o >>= 1) v += __shfl_down(v, o, 32);  // wave32
  return v;
}

// kept-mask + BCE element, accumulated per thread.
// log1pf(expf(-a)) -> __logf(1+__expf(-a)): native v_exp_f32/v_log_f32; the
// log1p advantage only matters for |x|>~9 where abs err <= 6e-8 -> negligible
// against a sum of ~8e5.
__device__ __forceinline__ void proc_elem(float x, float y, float& cnt, float& se) {
  bool kept = (y == 1.0f) ? (x <= LOGIT_07)
            : ((y == 0.0f) ? (x >= -LOGIT_07) : false);
  if (kept) {
    float elem = fmaxf(x, 0.0f) - x * y + __logf(1.0f + __expf(-fabsf(x)));
    se += elem;
    cnt += 1.0f;
  }
}

__device__ __forceinline__ size_t chunk_base(int c) {
  int b = c / CHUNKS_PER_BATCH;
  int o = c % CHUNKS_PER_BATCH;
  return (size_t)b * BATCH_STRIDE + (size_t)o * CHUNK;
}

// ---------------- K1: hot streaming pass (TDM double-buffered) ----------------
__global__ __launch_bounds__(256) void k1_main(const float* __restrict__ inp,
                                               const float* __restrict__ tgt,
                                               WS* ws) {
  __shared__ alignas(16) float sIn[2][CHUNK];
  __shared__ alignas(16) float sTg[2][CHUNK];
  __shared__ float redE[8], redC[8];

  const int tid  = threadIdx.x;
  const int wave = tid >> 5, lane = tid & 31;
  const int c0   = blockIdx.x * CHUNKS_PER_BLOCK;

  float cnt = 0.0f, se = 0.0f;

  if (wave == 0) {                       // wave0 drives the DMA pipeline
    size_t base = chunk_base(c0);
    tdm_load_f32(inp + base, &sIn[0][0], CHUNK);
    tdm_load_f32(tgt + base, &sTg[0][0], CHUNK);
  }

  for (int i = 0; i < CHUNKS_PER_BLOCK; ++i) {
    const int buf = i & 1;
    if (wave == 0) {
      if (i + 1 < CHUNKS_PER_BLOCK) {    // prefetch next chunk into other buffer
        size_t base = chunk_base(c0 + i + 1);
        tdm_load_f32(inp + base, &sIn[buf ^ 1][0], CHUNK);
        tdm_load_f32(tgt + base, &sTg[buf ^ 1][0], CHUNK);
        __builtin_amdgcn_s_wait_tensorcnt(2);  // in-order: current chunk done
      } else {
        __builtin_amdgcn_s_wait_tensorcnt(0);
      }
    }
    __syncthreads();                     // buffer `buf` visible to all waves
    #pragma unroll
    for (int k = 0; k < CHUNK / (256 * 4); ++k) {
      const int e = (k * 256 + tid) * 4;
      const float4 x4 = *(const float4*)&sIn[buf][e];
      const float4 y4 = *(const float4*)&sTg[buf][e];
      proc_elem(x4.x, y4.x, cnt, se);
      proc_elem(x4.y, y4.y, cnt, se);
      proc_elem(x4.z, y4.z, cnt, se);
      proc_elem(x4.w, y4.w, cnt, se);
    }
    __syncthreads();                     // done reading before buffer is reused
  }

  float e = wave_red(se), c = wave_red(cnt);
  if (lane == 0) { redE[wave] = e; redC[wave] = c; }
  __syncthreads();
  if (wave == 0) {
    float e2 = (lane < 8) ? redE[lane] : 0.0f;
    float c2 = (lane < 8) ? redC[lane] : 0.0f;
    e2 = wave_red(e2); c2 = wave_red(c2);
    if (lane == 0) {
      atomicAdd(&ws->sum_elem, (double)e2);
      atomicAdd(&ws->n_kept, c2);        // integer-valued, exact in f32 (< 2^24)
    }
  }
}

// ---------------- K2: fast-path finalize + flag ----------------
__global__ void k2_decide(WS* ws, float* out) {
  if (ws->n_kept >= 10000.0f) {
    ws->flag = 0u;
    out[0] = (float)((ws->sum_elem / (double)ws->n_kept) * 7.0);
  } else {
    ws->flag = 1u;
  }
}

// ---------------- K3: |x| histogram (only if fallback) ----------------
__global__ __launch_bounds__(256) void k3_hist(const float* __restrict__ inp, WS* ws) {
  if (ws->flag == 0u) return;
  __shared__ unsigned h[NBINS];
  const int tid = threadIdx.x;
  #pragma unroll
  for (int i = 0; i < NBINS / 256; ++i) h[tid + i * 256] = 0u;
  __syncthreads();
  const int nv4 = N_CH0 / 4;
  for (int v = blockIdx.x * blockDim.x + tid; v < nv4; v += gridDim.x * blockDim.x) {
    const int elem = v * 4;
    const size_t g = (size_t)(elem >> 18) * BATCH_STRIDE + (size_t)(elem & (HW - 1));
    const float4 x4 = *(const float4*)(inp + g);
    const float xs[4] = {x4.x, x4.y, x4.z, x4.w};
    #pragma unroll
    for (int j = 0; j < 4; ++j) {
      int b = (int)(fabsf(xs[j]) * BIN_SCALE);
      atomicAdd(&h[b > NBINS - 1 ? NBINS - 1 : b], 1u);
    }
  }
  __syncthreads();
  #pragma unroll
  for (int i = 0; i < NBINS / 256; ++i) {
    unsigned v = h[tid + i * 256];
    if (v) atomicAdd(&ws->hist[tid + i * 256], v);
  }
}

// ---------------- K4: histogram scan -> |x| threshold ----------------
__global__ void k4_thresh(WS* ws) {
  if (ws->flag == 0u) return;
  unsigned cum = 0; int bin = NBINS - 1;
  for (int i = 0; i < NBINS; ++i) { cum += ws->hist[i]; if (cum >= 10000u) { bin = i; break; } }
  ws->thresh = (bin >= NBINS - 1) ? 3.4e38f : (float)(bin + 1) / BIN_SCALE;
}

// ---------------- K5: union-mask re-pass (only if fallback) ----------------
__global__ __launch_bounds__(256) void k5_union(const float* __restrict__ inp,
                                                const float* __restrict__ tgt, WS* ws) {
  if (ws->flag == 0u) return;
  __shared__ float redE[8], redC[8];
  const int tid = threadIdx.x, wave = tid >> 5, lane = tid & 31;
  const float th = ws->thresh;
  float cnt = 0.0f, se = 0.0f;
  const int nv4 = N_CH0 / 4;
  for (int v = blockIdx.x * blockDim.x + tid; v < nv4; v += gridDim.x * blockDim.x) {
    const int elem = v * 4;
    const size_t g = (size_t)(elem >> 18) * BATCH_STRIDE + (size_t)(elem & (HW - 1));
    const float4 x4 = *(const float4*)(inp + g);
    const float4 y4 = *(const float4*)(tgt + g);
    const float xs[4] = {x4.x, x4.y, x4.z, x4.w};
    const float ys[4] = {y4.x, y4.y, y4.z, y4.w};
    #pragma unroll
    for (int j = 0; j < 4; ++j) {
      const float x = xs[j], y = ys[j];
      bool kept = (y == 1.0f) ? (x <= LOGIT_07)
                : ((y == 0.0f) ? (x >= -LOGIT_07) : false);
      kept = kept || (fabsf(x) < th);
      if (kept) {
        se += fmaxf(x, 0.0f) - x * y + __logf(1.0f + __expf(-fabsf(x)));
        cnt += 1.0f;
      }
    }
  }
  float e = wave_red(se), c = wave_red(cnt);
  if (lane == 0) { redE[wave] = e; redC[wave] = c; }
  __syncthreads();
  if (wave == 0) {
    float e2 = (lane < 8) ? redE[lane] : 0.0f;
    float c2 = (lane < 8) ? redC[lane] : 0.0f;
    e2 = wave_red(e2); c2 = wave_red(c2);
    if (lane == 0) {
      atomicAdd(&ws->sum_elem_u, (double)e2);
      atomicAdd(&ws->n_union, c2);
    }
  }
}

// ---------------- K6: fallback finalize ----------------
__global__ void k6_final(WS* ws, float* out) {
  if (ws->flag == 0u) return;
  out[0] = (float)((ws->sum_elem_u / (double)ws->n_union) * 7.0);
}

extern "C" void kernel_launch(void* const* d_in, const int* in_sizes, int n_in,
                              void* d_out, int out_size, void* d_ws, size_t ws_size,
                              hipStream_t stream) {
  (void)in_sizes; (void)n_in; (void)out_size; (void)ws_size;
  const float* inp = (const float*)d_in[0];
  const float* tgt = (const float*)d_in[1];
  float* out = (float*)d_out;
  WS* ws = (WS*)d_ws;

  (void)hipMemsetAsync(d_ws, 0, sizeof(WS), stream);
  k1_main<<<K1_BLOCKS, 256, 0, stream>>>(inp, tgt, ws);
  k2_decide<<<1, 1, 0, stream>>>(ws, out);
  k3_hist<<<256, 256, 0, stream>>>(inp, ws);
  k4_thresh<<<1, 1, 0, stream>>>(ws);
  k5_union<<<256, 256, 0, stream>>>(inp, tgt, ws);
  k6_final<<<1, 1, 0, stream>>>(ws, out);
}